// CenterNet_15427522527500
// MI455X (gfx1250) — compile-verified
//
#include <hip/hip_runtime.h>
#include <hip/hip_bf16.h>

// -------- problem constants (from reference) --------
#define M_ROWS   349184
#define C_CH     80
#define NPOS     2048
#define ALPHA    0.25f
#define CLAMP_LO 0.0001f
#define CLAMP_HI (1.0f - 0.0001f)

#define THREADS     256
#define NEG_BLOCKS  1024
#define REG_BLOCKS  512
// d_ws float layout: [0,1024) neg partials | [1024,1536) reg-loss partials
//                    [1536,2048) reg-w partials | [2048] pos sum
#define WS_REG_LOSS 1024
#define WS_REG_W    1536
#define WS_POS      2048

typedef __attribute__((ext_vector_type(2))) float v2f;
typedef __attribute__((ext_vector_type(8))) float v8f;

// Wave32 sum via V_WMMA_F32_16X16X4_F32: A(16x4) holds the 32 lane partials
// (lane L<16 -> A[L][0], lane L>=16 -> A[L-16][2]), B(4x16)=ones, C=0.
// D[m][n] = rowsum_m (independent of n). Per the C/D VGPR layout, summing the
// 8 D VGPRs gives rows 0-7 on lanes 0-15 and rows 8-15 on lanes 16-31, so the
// total is lane0 + lane16. Deterministic fp32 order every invocation.
__device__ __forceinline__ float wave_sum_wmma(float p) {
    v2f a; a[0] = p;    a[1] = 0.0f;
    v2f b; b[0] = 1.0f; b[1] = 1.0f;
    v8f c = {};
    c = __builtin_amdgcn_wmma_f32_16x16x4_f32(
            /*neg_a=*/false, a, /*neg_b=*/false, b,
            /*c_mod=*/(short)0, c, /*reuse_a=*/false, /*reuse_b=*/false);
    float s = c[0] + c[1] + c[2] + c[3] + c[4] + c[5] + c[6] + c[7];
    return __shfl(s, 0, 32) + __shfl(s, 16, 32);
}

// Deterministic block sum: 8 wave sums (WMMA) -> LDS -> fixed-order add.
__device__ __forceinline__ float block_sum(float v, float* lds) {
    float w = wave_sum_wmma(v);
    const int lane = threadIdx.x & 31;
    const int wid  = threadIdx.x >> 5;
    if (lane == 0) lds[wid] = w;
    __syncthreads();
    float total = 0.0f;
#pragma unroll
    for (int i = 0; i < THREADS / 32; ++i) total += lds[i];
    __syncthreads();
    return total;
}

__device__ __forceinline__ float sigmoid_clamped(float x) {
    float s = 1.0f / (1.0f + __expf(-x));
    return fminf(fmaxf(s, CLAMP_LO), CLAMP_HI);
}

// log(1-p) * p^2 * (1-hm)^4
__device__ __forceinline__ float neg_term(float x, float h) {
    float s  = sigmoid_clamped(x);
    float q  = 1.0f - h;
    float q2 = q * q;
    return __logf(1.0f - s) * (s * s) * (q2 * q2);
}

// ---------------- neg loss: stream M*C elements of two arrays ----------------
__global__ void __launch_bounds__(THREADS)
k_neg(const float* __restrict__ logits, const float* __restrict__ hms,
      float* __restrict__ partials) {
    __shared__ float lds[THREADS / 32];
    const size_t n4     = (size_t)M_ROWS * C_CH / 4;   // divisible: 80 % 4 == 0
    const size_t stride = (size_t)gridDim.x * blockDim.x;
    const float4* lp = (const float4*)logits;
    const float4* hp = (const float4*)hms;
    float acc = 0.0f;
    for (size_t i = (size_t)blockIdx.x * blockDim.x + threadIdx.x; i < n4; i += stride) {
        float4 x = lp[i];
        float4 h = hp[i];
        acc += neg_term(x.x, h.x) + neg_term(x.y, h.y) +
               neg_term(x.z, h.z) + neg_term(x.w, h.w);
    }
    float tot = block_sum(acc, lds);
    if (threadIdx.x == 0) partials[blockIdx.x] = tot;
}

// ---------------- pos loss: 2048 gathered elements ----------------
__global__ void __launch_bounds__(THREADS)
k_pos(const float* __restrict__ logits, const int* __restrict__ pos_inds,
      const int* __restrict__ labels, float* __restrict__ out_pos) {
    __shared__ float lds[THREADS / 32];
    float acc = 0.0f;
    for (int i = threadIdx.x; i < NPOS; i += THREADS) {
        int r = pos_inds[i];
        int c = labels[i];
        float s  = sigmoid_clamped(logits[(size_t)r * C_CH + c]);
        float om = 1.0f - s;
        acc += __logf(s) * om * om;
    }
    float tot = block_sum(acc, lds);
    if (threadIdx.x == 0) *out_pos = tot;
}

// ---------------- reg loss: GIoU over M rows of 4 ----------------
__global__ void __launch_bounds__(THREADS)
k_reg(const float4* __restrict__ pred, const float4* __restrict__ targ,
      float* __restrict__ loss_part, float* __restrict__ w_part) {
    __shared__ float lds[THREADS / 32];
    const size_t stride = (size_t)gridDim.x * blockDim.x;
    float lacc = 0.0f, wacc = 0.0f;
    for (size_t i = (size_t)blockIdx.x * blockDim.x + threadIdx.x; i < (size_t)M_ROWS; i += stride) {
        float4 p = pred[i];
        float4 t = targ[i];
        float mx = fmaxf(fmaxf(t.x, t.y), fmaxf(t.z, t.w));
        float w  = (mx >= 0.0f) ? 1.0f : 0.0f;
        // safe targets (1.0 where invalid) keep math finite
        float tl = (mx >= 0.0f) ? t.x : 1.0f;
        float tt = (mx >= 0.0f) ? t.y : 1.0f;
        float tr = (mx >= 0.0f) ? t.z : 1.0f;
        float tb = (mx >= 0.0f) ? t.w : 1.0f;
        float pl = p.x, pt = p.y, pr = p.z, pb = p.w;

        float target_area = (tl + tr) * (tt + tb);
        float pred_area   = (pl + pr) * (pt + pb);
        float w_int = fminf(pl, tl) + fminf(pr, tr);
        float h_int = fminf(pb, tb) + fminf(pt, tt);
        float g_w   = fmaxf(pl, tl) + fmaxf(pr, tr);
        float g_h   = fmaxf(pb, tb) + fmaxf(pt, tt);
        float ac    = g_w * g_h;
        float ai    = w_int * h_int;
        float au    = target_area + pred_area - ai;
        float ious  = (ai + 1.0f) / (au + 1.0f);
        float gious = ious - (ac - au) / (ac + 1e-7f);
        lacc += (1.0f - gious) * w;
        wacc += w;
    }
    float lt = block_sum(lacc, lds);
    float wt = block_sum(wacc, lds);
    if (threadIdx.x == 0) {
        loss_part[blockIdx.x] = lt;
        w_part[blockIdx.x]    = wt;
    }
}

// ---------------- final combine (fixed order => deterministic) ----------------
__global__ void __launch_bounds__(THREADS)
k_final(const float* __restrict__ ws, float* __restrict__ out) {
    __shared__ float lds[THREADS / 32];
    float a = 0.0f;
    for (int i = threadIdx.x; i < NEG_BLOCKS; i += THREADS) a += ws[i];
    float neg_sum = block_sum(a, lds);
    float b = 0.0f;
    for (int i = threadIdx.x; i < REG_BLOCKS; i += THREADS) b += ws[WS_REG_LOSS + i];
    float reg_sum = block_sum(b, lds);
    float c = 0.0f;
    for (int i = threadIdx.x; i < REG_BLOCKS; i += THREADS) c += ws[WS_REG_W + i];
    float w_sum = block_sum(c, lds);
    if (threadIdx.x == 0) {
        float pos_sum = ws[WS_POS];
        const float inv_npos = 1.0f / (float)NPOS;
        out[0] = (-ALPHA)          * pos_sum * inv_npos;          // POS_W = 1
        out[1] = (-(1.0f - ALPHA)) * neg_sum * inv_npos;          // NEG_W = 1
        out[2] = 2.0f * reg_sum / fmaxf(w_sum, 1.0f);             // REG_W = 2
    }
}

extern "C" void kernel_launch(void* const* d_in, const int* in_sizes, int n_in,
                              void* d_out, int out_size, void* d_ws, size_t ws_size,
                              hipStream_t stream) {
    const float* logits   = (const float*)d_in[0];
    const float* hms      = (const float*)d_in[1];
    const float* reg_pred = (const float*)d_in[2];
    const float* reg_targ = (const float*)d_in[3];
    const int*   pos_inds = (const int*)d_in[4];
    const int*   labels   = (const int*)d_in[5];
    float* out = (float*)d_out;
    float* ws  = (float*)d_ws;

    k_neg<<<NEG_BLOCKS, THREADS, 0, stream>>>(logits, hms, ws);
    k_reg<<<REG_BLOCKS, THREADS, 0, stream>>>((const float4*)reg_pred,
                                              (const float4*)reg_targ,
                                              ws + WS_REG_LOSS, ws + WS_REG_W);
    k_pos<<<1, THREADS, 0, stream>>>(logits, pos_inds, labels, ws + WS_POS);
    k_final<<<1, THREADS, 0, stream>>>(ws, out);
}